// TealActor_60559038873968
// MI455X (gfx1250) — compile-verified
//
#include <hip/hip_runtime.h>
#include <math.h>

#define N_LINK   4000
#define N_FLOW   30000
#define NUM_PATH 4
#define N_PATH   120000
#define E_TOPO   32000
#define E_LP     720000
#define HEADS    4
#define DHEAD    32
#define HID      128

typedef __attribute__((ext_vector_type(2))) float v2f;
typedef __attribute__((ext_vector_type(8))) float v8f;

// ---- workspace layout (float offsets) ----
#define WS_HS2   0        // 16000  : hs2[N_LINK*4]   (topo layer-2 head values)
#define WS_XLINK 16000    // 4000   : x_link[N_LINK]
#define WS_UVEC  20000    // 640    : u_ls, v_ld, u_fs, v_fd, w_self (5 x 128)
#define WS_COEF  20640    // 16     : c1s[4], c1d[4], c_ls[4], c_ld[4]
#define WS_SCORE 20656    // 120000 : score[N_PATH]

__device__ __forceinline__ float lrelu02(float x) { return x >= 0.f ? x : 0.2f * x; }
__device__ __forceinline__ float eluf(float x)    { return x > 0.f ? x : (expf(x) - 1.f); }

// ---- per-head attention coefficients for topo layer 1 (rank-1 collapse) ----
__global__ void k_coef1(const float* __restrict__ Wt1, const float* __restrict__ a1s,
                        const float* __restrict__ a1d, float* __restrict__ coef) {
    int t = threadIdx.x;
    if (t < 8) {
        int h = t & 3;
        bool dstp = t >= 4;
        const float* a = dstp ? a1d : a1s;
        float acc = 0.f;
        for (int k = 0; k < DHEAD; ++k) acc += Wt1[h * DHEAD + k] * a[h * DHEAD + k];
        coef[(dstp ? 4 : 0) + h] = acc;
    }
}

// ---- topo GAT layer 1 fused with ELU and @Wt2 -> hs2[N_LINK,4] ----
// One wave per dst link: lanes scan the edge list strided (int4-coalesced),
// accumulate per-head softmax sums, then a fixed shfl-tree reduction
// (deterministic, no atomics). Softmax max pass dropped (shift invariance).
__global__ void k_topo1(const float* __restrict__ cap, const int* __restrict__ tsrc,
                        const int* __restrict__ tdst, const float* __restrict__ Wt1,
                        const float* __restrict__ Wt2, const float* __restrict__ coef,
                        float* __restrict__ hs2) {
    int wave = threadIdx.x >> 5;
    int lane = threadIdx.x & 31;
    int d = blockIdx.x * 4 + wave;            // N_LINK % 4 == 0
    float c1s[4], c1d[4];
#pragma unroll
    for (int h = 0; h < 4; ++h) { c1s[h] = coef[h]; c1d[h] = coef[4 + h]; }
    float cd = cap[d];
    float wsum[4] = {0.f, 0.f, 0.f, 0.f}, wx[4] = {0.f, 0.f, 0.f, 0.f};
    const int4* tdst4 = (const int4*)tdst;
    for (int v4 = lane; v4 < E_TOPO / 4; v4 += 32) {
        int4 dd = tdst4[v4];
        int dv[4] = {dd.x, dd.y, dd.z, dd.w};
#pragma unroll
        for (int c = 0; c < 4; ++c) {
            if (dv[c] == d) {
                float cs = cap[tsrc[v4 * 4 + c]];
#pragma unroll
                for (int h = 0; h < 4; ++h) {
                    float ev = lrelu02(cs * c1s[h] + cd * c1d[h]);
                    float w = expf(ev);
                    wsum[h] += w;
                    wx[h] += w * cs;
                }
            }
        }
    }
#pragma unroll
    for (int off = 16; off > 0; off >>= 1) {
#pragma unroll
        for (int h = 0; h < 4; ++h) {
            wsum[h] += __shfl_down(wsum[h], off);
            wx[h]   += __shfl_down(wx[h], off);
        }
    }
    float s[4];
#pragma unroll
    for (int h = 0; h < 4; ++h) {
        float tw = __shfl(wsum[h], 0);
        float tx = __shfl(wx[h], 0);
        s[h] = tx / (tw + 1e-9f);
    }
    // j-loop: lane handles j = lane + 32q  (j>>5 == q); Wt2 row as float4
    float acc[4] = {0.f, 0.f, 0.f, 0.f};
    const float4* Wt2r = (const float4*)Wt2;
#pragma unroll
    for (int q = 0; q < 4; ++q) {
        int j = lane + 32 * q;
        float v = eluf(s[q] * Wt1[j]);        // h1[d,j] never materialized
        float4 w4 = Wt2r[j];
        acc[0] += v * w4.x; acc[1] += v * w4.y;
        acc[2] += v * w4.z; acc[3] += v * w4.w;
    }
#pragma unroll
    for (int off = 16; off > 0; off >>= 1) {
#pragma unroll
        for (int h2 = 0; h2 < 4; ++h2) acc[h2] += __shfl_down(acc[h2], off);
    }
    if (lane == 0) {
#pragma unroll
        for (int h2 = 0; h2 < 4; ++h2) hs2[d * 4 + h2] = acc[h2];
    }
}

// ---- topo GAT layer 2 (Dh=1 heads) + mean over heads -> x_link[N_LINK] ----
__global__ void k_topo2(const int* __restrict__ tsrc, const int* __restrict__ tdst,
                        const float* __restrict__ a2s, const float* __restrict__ a2d,
                        const float* __restrict__ hs2, float* __restrict__ x_link) {
    int wave = threadIdx.x >> 5;
    int lane = threadIdx.x & 31;
    int d = blockIdx.x * 4 + wave;
    float hd[4], as[4], ad[4];
#pragma unroll
    for (int h = 0; h < 4; ++h) { hd[h] = hs2[d * 4 + h]; as[h] = a2s[h]; ad[h] = a2d[h]; }
    float wsum[4] = {0.f, 0.f, 0.f, 0.f}, wx[4] = {0.f, 0.f, 0.f, 0.f};
    const int4* tdst4 = (const int4*)tdst;
    for (int v4 = lane; v4 < E_TOPO / 4; v4 += 32) {
        int4 dd = tdst4[v4];
        int dv[4] = {dd.x, dd.y, dd.z, dd.w};
#pragma unroll
        for (int c = 0; c < 4; ++c) {
            if (dv[c] == d) {
                int si = tsrc[v4 * 4 + c];
#pragma unroll
                for (int h = 0; h < 4; ++h) {
                    float hv = hs2[si * 4 + h];
                    float ev = lrelu02(hv * as[h] + hd[h] * ad[h]);
                    float w = expf(ev);
                    wsum[h] += w;
                    wx[h] += w * hv;
                }
            }
        }
    }
#pragma unroll
    for (int off = 16; off > 0; off >>= 1) {
#pragma unroll
        for (int h = 0; h < 4; ++h) {
            wsum[h] += __shfl_down(wsum[h], off);
            wx[h]   += __shfl_down(wx[h], off);
        }
    }
    if (lane == 0) {
        float x = 0.f;
#pragma unroll
        for (int h = 0; h < 4; ++h) x += wx[h] / (wsum[h] + 1e-9f);
        x_link[d] = 0.25f * x;
    }
}

// ---- 5 x (1x128 @ 128x128) vector-matrix products on the WMMA pipe ----
// Each block computes one product; wave w computes output columns [16w,16w+16)
// via a K-loop of 32 V_WMMA_F32_16X16X4_F32 ops. A rows are replicated copies
// of the 1x128 row vector; row 0 of the 16x16 C tile is the result.
__global__ void k_wmma_vecmat(const float* __restrict__ Wl, const float* __restrict__ Wp,
                              const float* __restrict__ Wf,
                              const float* __restrict__ Wl2p_s, const float* __restrict__ Wl2p_d,
                              const float* __restrict__ Wf2p_s, const float* __restrict__ Wf2p_d,
                              const float* __restrict__ Wself, float* __restrict__ uvec) {
    const float* row;
    const float* mat;
    switch (blockIdx.x) {
        case 0:  row = Wl; mat = Wl2p_s; break;   // u_ls
        case 1:  row = Wp; mat = Wl2p_d; break;   // v_ld
        case 2:  row = Wf; mat = Wf2p_s; break;   // u_fs
        case 3:  row = Wp; mat = Wf2p_d; break;   // v_fd (kept for generality)
        default: row = Wp; mat = Wself;  break;   // w_self
    }
    int wave = threadIdx.x >> 5;
    int lane = threadIdx.x & 31;
    int half = lane >> 4;        // ISA 7.12.2: lanes 16-31 hold K+2 / K+3
    int l16  = lane & 15;
    int col0 = wave * 16;
    v8f acc = {};
    for (int kk = 0; kk < 32; ++kk) {
        int k0 = 4 * kk + 2 * half;
        v2f a, b;
        a.x = row[k0];
        a.y = row[k0 + 1];
        b.x = mat[(k0) * HID + col0 + l16];
        b.y = mat[(k0 + 1) * HID + col0 + l16];
        acc = __builtin_amdgcn_wmma_f32_16x16x4_f32(false, a, false, b,
                                                    (short)0, acc, false, false);
    }
    // C/D layout: VGPR0, lanes 0-15 => M=0, N=lane
    if (lane < 16) uvec[blockIdx.x * HID + col0 + lane] = acc[0];
}

// ---- per-head coefficients for the link->path attention ----
__global__ void k_coef2(const float* __restrict__ a_ls, const float* __restrict__ a_ld,
                        const float* __restrict__ uvec, float* __restrict__ coef) {
    int t = threadIdx.x;
    if (t < 8) {
        int h = t & 3;
        bool dstp = t >= 4;
        const float* a = dstp ? a_ld : a_ls;
        const float* u = dstp ? (uvec + HID) : uvec;   // v_ld : u_ls
        float acc = 0.f;
        for (int k = 0; k < DHEAD; ++k) acc += u[h * DHEAD + k] * a[h * DHEAD + k];
        coef[8 + (dstp ? 4 : 0) + h] = acc;
    }
}

// ---- fused: link->path segmented-softmax GAT (lp_dst sorted => binary search
// + contiguous run, deterministic, no atomics) + flow->path (single-edge
// softmax = const) + Wself + ReLU + Wout -> score[N_PATH] ----
__global__ void k_path_score(const float* __restrict__ path_feat, const float* __restrict__ flow_feat,
                             const float* __restrict__ edge_w, const int* __restrict__ lp_src,
                             const int* __restrict__ lp_dst, const int* __restrict__ fp_src,
                             const float* __restrict__ x_link, const float* __restrict__ uvec,
                             const float* __restrict__ coef, const float* __restrict__ Wout,
                             const float* __restrict__ bout, float* __restrict__ score) {
    int d = blockIdx.x * blockDim.x + threadIdx.x;
    if (d >= N_PATH) return;
    float cls[4], cld[4];
#pragma unroll
    for (int h = 0; h < 4; ++h) { cls[h] = coef[8 + h]; cld[h] = coef[12 + h]; }
    float pf = path_feat[d];
    // lower_bound(lp_dst, d)
    int lo = 0, hi = E_LP;
    while (lo < hi) {
        int mid = (lo + hi) >> 1;
        if (lp_dst[mid] < d) lo = mid + 1; else hi = mid;
    }
    float wsum[4] = {0.f, 0.f, 0.f, 0.f}, wx[4] = {0.f, 0.f, 0.f, 0.f};
    for (int e = lo; e < E_LP; ++e) {
        if (lp_dst[e] != d) break;
        float xl = x_link[lp_src[e]];
        float ew = edge_w[e];
#pragma unroll
        for (int h = 0; h < 4; ++h) {
            float ev = lrelu02(xl * cls[h] + pf * cld[h]);
            float w = expf(ev) * ew;
            wsum[h] += w;
            wx[h] += w * xl;
        }
    }
    float Sl[4];
#pragma unroll
    for (int h = 0; h < 4; ++h) Sl[h] = wx[h] / (wsum[h] + 1e-9f);
    // flow->path: exactly one edge per dst -> alpha = 1/(1+1e-9)
    float ff = flow_feat[fp_src[d]] * (1.0f / (1.0f + 1e-9f));
    const float* uls   = uvec;             // Wl @ Wl2p_s
    const float* ufs   = uvec + 2 * HID;   // Wf @ Wf2p_s
    const float* wself = uvec + 4 * HID;   // Wp @ Wself
    float acc = 0.f;
    for (int j = 0; j < HID; ++j) {
        float t = pf * wself[j] + Sl[j >> 5] * uls[j] + ff * ufs[j];
        t = t > 0.f ? t : 0.f;
        acc += t * Wout[j];
    }
    score[d] = acc + bout[0];
}

// ---- mean head: score.reshape(N_FLOW,4) @ Wm.T + bm ----
__global__ void k_mean(const float* __restrict__ score, const float* __restrict__ Wm,
                       const float* __restrict__ bm, float* __restrict__ out) {
    int t = blockIdx.x * blockDim.x + threadIdx.x;
    if (t >= N_PATH) return;
    int f = t >> 2, p = t & 3;
    float acc = bm[p];
#pragma unroll
    for (int q = 0; q < 4; ++q) acc += score[f * 4 + q] * Wm[p * 4 + q];
    out[t] = acc;
}

extern "C" void kernel_launch(void* const* d_in, const int* in_sizes, int n_in,
                              void* d_out, int out_size, void* d_ws, size_t ws_size,
                              hipStream_t stream) {
    (void)in_sizes; (void)n_in; (void)out_size; (void)ws_size;
    const float* capacity  = (const float*)d_in[0];
    const float* flow_feat = (const float*)d_in[1];
    const float* path_feat = (const float*)d_in[2];
    const float* edge_w    = (const float*)d_in[3];
    const int*   topo_src  = (const int*)d_in[4];
    const int*   topo_dst  = (const int*)d_in[5];
    const int*   fp_src    = (const int*)d_in[6];
    /* d_in[7] fp_dst == arange, unused */
    const int*   lp_src    = (const int*)d_in[8];
    const int*   lp_dst    = (const int*)d_in[9];
    const float* Wt1    = (const float*)d_in[10];
    const float* a1s    = (const float*)d_in[11];
    const float* a1d    = (const float*)d_in[12];
    const float* Wt2    = (const float*)d_in[13];
    const float* a2s    = (const float*)d_in[14];
    const float* a2d    = (const float*)d_in[15];
    const float* Wf     = (const float*)d_in[16];
    const float* Wp     = (const float*)d_in[17];
    const float* Wl     = (const float*)d_in[18];
    const float* Wl2p_s = (const float*)d_in[19];
    const float* Wl2p_d = (const float*)d_in[20];
    const float* a_ls   = (const float*)d_in[21];
    const float* a_ld   = (const float*)d_in[22];
    const float* Wf2p_s = (const float*)d_in[23];
    const float* Wf2p_d = (const float*)d_in[24];
    /* a_fs (25), a_fd (26): single-edge softmax -> alpha const, unused */
    const float* Wself  = (const float*)d_in[27];
    const float* Wout   = (const float*)d_in[28];
    const float* bout   = (const float*)d_in[29];
    const float* Wm     = (const float*)d_in[30];
    const float* bm     = (const float*)d_in[31];

    float* ws     = (float*)d_ws;
    float* hs2    = ws + WS_HS2;
    float* x_link = ws + WS_XLINK;
    float* uvec   = ws + WS_UVEC;
    float* coef   = ws + WS_COEF;
    float* score  = ws + WS_SCORE;
    float* out    = (float*)d_out;

    k_coef1<<<1, 32, 0, stream>>>(Wt1, a1s, a1d, coef);
    // one wave per dst link: 4 waves (128 thr) per block, N_LINK/4 blocks
    k_topo1<<<N_LINK / 4, 128, 0, stream>>>(capacity, topo_src, topo_dst,
                                            Wt1, Wt2, coef, hs2);
    k_topo2<<<N_LINK / 4, 128, 0, stream>>>(topo_src, topo_dst, a2s, a2d,
                                            hs2, x_link);
    k_wmma_vecmat<<<5, 256, 0, stream>>>(Wl, Wp, Wf, Wl2p_s, Wl2p_d,
                                         Wf2p_s, Wf2p_d, Wself, uvec);
    k_coef2<<<1, 32, 0, stream>>>(a_ls, a_ld, uvec, coef);
    k_path_score<<<(N_PATH + 127) / 128, 128, 0, stream>>>(path_feat, flow_feat, edge_w,
                                                           lp_src, lp_dst, fp_src, x_link,
                                                           uvec, coef, Wout, bout, score);
    k_mean<<<(N_PATH + 255) / 256, 256, 0, stream>>>(score, Wm, bm, out);
}